// Averager_47115791237167
// MI455X (gfx1250) — compile-verified
//
#include <hip/hip_runtime.h>
#include <hip/hip_bf16.h>

typedef __attribute__((ext_vector_type(2))) float v2f;
typedef __attribute__((ext_vector_type(8))) float v8f;

#define HAVE_WMMA_F32X4 __has_builtin(__builtin_amdgcn_wmma_f32_16x16x4_f32)

// Accumulate 16 row-sums (rows rowBase..rowBase+15) of kTotal elements each,
// scaled by `scale`, using V_WMMA_F32_16X16X4_F32 chains with an all-`scale`
// B matrix: D[m][n] = sum_k A[m][k]*scale. Result column: lane 0 holds rows
// rowBase..+7 in acc[0..7], lane 16 holds rows rowBase+8..+15.
// A 16x4 f32 VGPR layout (ISA 7.12.2): lanes 0-15: VGPR0=K0,VGPR1=K1;
// lanes 16-31: VGPR0=K2,VGPR1=K3  -> per lane one float2 LDS read per step.
__device__ __forceinline__ v8f wmma_rowsum16_acc(const float* Lp, int rowBase,
                                                 int rowStride, int kTotal,
                                                 float scale, v8f acc) {
#if HAVE_WMMA_F32X4
  const int lane = threadIdx.x & 31;
  const int rr = lane & 15;
  const int h = (lane >> 4) & 1;
  const float* rp = Lp + (rowBase + rr) * rowStride + 2 * h;
  v2f bb; bb[0] = scale; bb[1] = scale;
  for (int g = 0; g < kTotal; g += 4) {
    v2f aa = *(const v2f*)(rp + g);
    acc = __builtin_amdgcn_wmma_f32_16x16x4_f32(false, aa, false, bb,
                                                (short)0, acc, false, false);
  }
#else
  const int lane = threadIdx.x & 31;
  const int h = (lane >> 4) & 1;
#pragma unroll
  for (int r = 0; r < 8; ++r) {
    const float* rp = Lp + (rowBase + h * 8 + r) * rowStride;
    float s = 0.f;
    for (int k = 0; k < kTotal; ++k) s += rp[k];
    acc[r] += s * scale;
  }
#endif
  return acc;
}

// ---------------- Level 0: pure digit-permutation copy ---------------------
// out[Zh Yh Xh | X4 Y4 Z4] = slab0[Y*65536 + Z*256 + X]
__global__ __launch_bounds__(256) void k_l0(const float* __restrict__ x,
                                            float* __restrict__ out) {
  __shared__ __align__(16) float L[4160];  // 4096 + swizzle pad
  const int t = threadIdx.x;
  const int tile = blockIdx.x & 4095;      // (Zh<<6)|Yh
  const int b = blockIdx.x >> 12;
  const int Zh = tile >> 6, Yh = tile & 63;
  const float* src = x + (size_t)b * 67108864u + (size_t)Yh * 262144u +
                     (size_t)Zh * 1024u;
#pragma unroll
  for (int v = 0; v < 4; ++v) {
    int li = (v * 256 + t) << 2;           // in-tile: Y4*1024 + Z4*256 + X
    int Y4 = li >> 10, Z4 = (li >> 8) & 3, X = li & 255;
    float4 d = *(const float4*)(src + Y4 * 65536 + Z4 * 256 + X);
    *(float4*)&L[li + ((li >> 8) << 2)] = d;
  }
  __syncthreads();
  float* ot = out + (size_t)b * 67108864u + ((size_t)tile << 12);
#pragma unroll
  for (int v = 0; v < 4; ++v) {
    int u = v * 256 + t;                   // (Xh,X4,Y4)
    int Y4 = u & 3, X4 = (u >> 2) & 3, Xh = u >> 4;
    int lib = Y4 * 1024 + Xh * 4 + X4;
    float4 o;
    int l0 = lib;       o.x = L[l0 + ((l0 >> 8) << 2)];
    int l1 = lib + 256; o.y = L[l1 + ((l1 >> 8) << 2)];
    int l2 = lib + 512; o.z = L[l2 + ((l2 >> 8) << 2)];
    int l3 = lib + 768; o.w = L[l3 + ((l3 >> 8) << 2)];
    *(float4*)(ot + (u << 2)) = o;
  }
}

// ---------------- Level 1 reduce: D1[y'(c2c3b1)][z'(b2b3p1)][x'(a1a2a3)] ----
// mean over (c1,a4,q1). One WG per (b, c2c3b1b2b3); rows=(a1a2a3,p1)=256,
// K=64 summed via two LDS passes (c1 pairs) of K=32.
__global__ __launch_bounds__(256) void k_l1a(const float* __restrict__ x,
                                             float* __restrict__ D1) {
  __shared__ __align__(16) float L[256 * 36];
  const int t = threadIdx.x;
  const int j = blockIdx.x & 1023;         // (c2,c3,b1,b2,b3)
  const int b = blockIdx.x >> 10;
  const float* base = x + (size_t)b * 67108864u + 16777216u +
      (size_t)(j >> 8) * 1048576u + (size_t)((j >> 6) & 3) * 262144u +
      (size_t)((j >> 4) & 3) * 65536u + (size_t)((j >> 2) & 3) * 16384u +
      (size_t)(j & 3) * 4096u;
  const int wave = t >> 5;
  v8f accA = {}; v8f accB = {};
  for (int p = 0; p < 2; ++p) {
    if (p) __syncthreads();
#pragma unroll
    for (int c1h = 0; c1h < 2; ++c1h) {
      const float* cb = base + (size_t)(p * 2 + c1h) * 4194304u;
#pragma unroll
      for (int v = 0; v < 4; ++v) {
        int idx = (v * 256 + t) << 2;      // a1a2a3 a4 p1 q1
        int row = ((idx >> 6) << 2) | ((idx >> 2) & 3);
        int kk = (c1h << 4) | ((idx & 0x30) >> 2);  // (c1h,a4,q1)
        *(float4*)&L[row * 36 + kk] = *(const float4*)(cb + idx);
      }
    }
    __syncthreads();
    accA = wmma_rowsum16_acc(L, wave * 32,      36, 32, 1.0f / 64.0f, accA);
    accB = wmma_rowsum16_acc(L, wave * 32 + 16, 36, 32, 1.0f / 64.0f, accB);
  }
  const int lane = t & 31;
  const int yp = j >> 4;
  if (lane == 0 || lane == 16) {
    const int rsel = (lane == 0) ? 0 : 8;
#pragma unroll
    for (int r = 0; r < 8; ++r) {
      int rowA = wave * 32 + rsel + r;
      int zpA = ((j & 15) << 2) | (rowA & 3);
      D1[(((b << 12) | (yp << 6) | zpA) << 6) | (rowA >> 2)] = accA[r];
      int rowB = wave * 32 + 16 + rsel + r;
      int zpB = ((j & 15) << 2) | (rowB & 3);
      D1[(((b << 12) | (yp << 6) | zpB) << 6) | (rowB >> 2)] = accB[r];
    }
  }
}

// Level 1 broadcast: out1[X,Y,Z] = D1[Y2Y3Y4][Z2Z3Z4][X2X3X4]
__global__ __launch_bounds__(256) void k_l1b(const float* __restrict__ D1,
                                             float* __restrict__ out) {
  __shared__ float L[16 * 68];
  const int t = threadIdx.x;
  const int tile = blockIdx.x & 4095;
  const int b = blockIdx.x >> 12;
  const int Zh = tile >> 6, Yh = tile & 63;
  const int xp0 = t & 63;
  for (int ss = t >> 6; ss < 16; ss += 4) {
    int Y4 = ss >> 2, Z4 = ss & 3;
    int yp = ((Yh & 15) << 2) | Y4;
    int zp = ((Zh & 15) << 2) | Z4;
    L[ss * 68 + xp0] = D1[(((b << 12) | (yp << 6) | zp) << 6) | xp0];
  }
  __syncthreads();
  float* ot = out + (size_t)b * 67108864u + 16777216u + ((size_t)tile << 12);
#pragma unroll
  for (int v = 0; v < 4; ++v) {
    int u = v * 256 + t;
    int Y4 = u & 3, X4 = (u >> 2) & 3, Xh = u >> 4;
    int xp = ((Xh & 15) << 2) | X4;
    float4 o;
    o.x = L[((Y4 << 2) | 0) * 68 + xp];
    o.y = L[((Y4 << 2) | 1) * 68 + xp];
    o.z = L[((Y4 << 2) | 2) * 68 + xp];
    o.w = L[((Y4 << 2) | 3) * 68 + xp];
    *(float4*)(ot + (u << 2)) = o;
  }
}

// ---------------- Level 2 reduce stage 1: partials over (a3,a4,q1,q2) ------
// WG per (b, c1c2b1b2a1) chunk; rows=(a2,p1,p2)=64, K=256 via 2 passes of 128.
__global__ __launch_bounds__(256) void k_l2a(const float* __restrict__ x,
                                             float* __restrict__ P2) {
  __shared__ __align__(16) float L[64 * 132];
  const int t = threadIdx.x;
  const int j = blockIdx.x & 1023;         // (c1,c2,b1,b2,a1)
  const int b = blockIdx.x >> 10;
  const float* base = x + (size_t)b * 67108864u + 33554432u +
      (size_t)(j >> 8) * 4194304u + (size_t)((j >> 6) & 3) * 1048576u +
      (size_t)((j >> 4) & 3) * 262144u + (size_t)((j >> 2) & 3) * 65536u +
      (size_t)(j & 3) * 16384u;
  const int wave = t >> 5;
  v8f acc = {};
  for (int p = 0; p < 2; ++p) {
    if (p) __syncthreads();
#pragma unroll
    for (int v = 0; v < 8; ++v) {
      int e = (v * 256 + t) << 2;
      int a2 = (e >> 11) & 3, a3lo = (e >> 10) & 1;
      int idx = (a2 << 12) | (p << 11) | (a3lo << 10) | (e & 1023);
      int row = (a2 << 4) | (((e >> 6) & 3) << 2) | ((e >> 4) & 3);
      int kk = (a3lo << 6) | (((e >> 8) & 3) << 4) | (((e >> 2) & 3) << 2);
      *(float4*)&L[row * 132 + kk] = *(const float4*)(base + idx);
    }
    __syncthreads();
    if (wave < 4)
      acc = wmma_rowsum16_acc(L, wave * 16, 132, 128, 1.0f / 4096.0f, acc);
  }
  if (wave < 4) {
    const int lane = t & 31;
    if (lane == 0 || lane == 16) {
      const int rsel = (lane == 0) ? 0 : 8;
      float* dst = P2 + ((((size_t)b << 10) | (unsigned)j) << 6);
#pragma unroll
      for (int r = 0; r < 8; ++r) dst[wave * 16 + rsel + r] = acc[r];
    }
  }
}

// Level 2 reduce stage 2: sum 16 (c1,c2) chunk partials -> D2[y'][z'][x']
__global__ __launch_bounds__(256) void k_l2a2(const float* __restrict__ P2,
                                              float* __restrict__ D2) {
  int tt = blockIdx.x * 256 + threadIdx.x;  // (b,yp,zp,xp)
  int b = tt >> 12;
  int yp = (tt >> 8) & 15, zp = (tt >> 4) & 15, xp = tt & 15;
  int b1 = yp >> 2, b2 = yp & 3, p1 = zp >> 2, p2 = zp & 3;
  int a1 = xp >> 2, a2 = xp & 3;
  int row = (a2 << 4) | (p1 << 2) | p2;
  float s = 0.f;
#pragma unroll
  for (int cc = 0; cc < 16; ++cc) {
    int chunk = (cc << 6) | (b1 << 4) | (b2 << 2) | a1;
    s += P2[((((size_t)b << 10) | (unsigned)chunk) << 6) | (unsigned)row];
  }
  D2[tt] = s;
}

// Level 2 broadcast: out2[X,Y,Z] = D2[Y3Y4][Z3Z4][X3X4]
__global__ __launch_bounds__(256) void k_l2b(const float* __restrict__ D2,
                                             float* __restrict__ out) {
  __shared__ float L[16 * 20];
  const int t = threadIdx.x;
  const int tile = blockIdx.x & 4095;
  const int b = blockIdx.x >> 12;
  const int Zh = tile >> 6, Yh = tile & 63;
  {
    int pair = t >> 4, xp = t & 15;
    int yp = ((Yh & 3) << 2) | (pair >> 2);
    int zp = ((Zh & 3) << 2) | (pair & 3);
    L[pair * 20 + xp] = D2[(b << 12) | (yp << 8) | (zp << 4) | xp];
  }
  __syncthreads();
  float* ot = out + (size_t)b * 67108864u + 33554432u + ((size_t)tile << 12);
#pragma unroll
  for (int v = 0; v < 4; ++v) {
    int u = v * 256 + t;
    int Y4 = u & 3, X4 = (u >> 2) & 3, Xh = u >> 4;
    int xp = ((Xh & 3) << 2) | X4;
    float4 o;
    o.x = L[((Y4 << 2) | 0) * 20 + xp];
    o.y = L[((Y4 << 2) | 1) * 20 + xp];
    o.z = L[((Y4 << 2) | 2) * 20 + xp];
    o.w = L[((Y4 << 2) | 3) * 20 + xp];
    *(float4*)(ot + (u << 2)) = o;
  }
}

// ---------------- Level 3 reduce stage 1: WG per (b, c1b1a1a2a3) chunk -----
// rows=(p2,p3)=16, K=1024 over (a4,p1,q1,q2,q3) via 2 passes of 512.
__global__ __launch_bounds__(256) void k_l3a(const float* __restrict__ x,
                                             float* __restrict__ P3) {
  __shared__ __align__(16) float L[16 * 516];
  const int t = threadIdx.x;
  const int j = blockIdx.x & 1023;         // (c1,b1,a1,a2,a3)
  const int b = blockIdx.x >> 10;
  const float* base = x + (size_t)b * 67108864u + 50331648u +
      (size_t)(j >> 8) * 4194304u + (size_t)((j >> 6) & 3) * 1048576u +
      (size_t)((j >> 4) & 3) * 262144u + (size_t)((j >> 2) & 3) * 65536u +
      (size_t)(j & 3) * 16384u;
  const int wave = t >> 5;
  v8f acc = {};
  for (int p = 0; p < 2; ++p) {
    if (p) __syncthreads();
#pragma unroll
    for (int v = 0; v < 8; ++v) {
      int e = (v * 256 + t) << 2;
      int idx = (p << 13) | e;
      int row = (((e >> 8) & 3) << 2) | ((e >> 6) & 3);
      int kk = (((e >> 12) & 1) << 8) | (((e >> 10) & 3) << 6) |
               (((e >> 4) & 3) << 4) | (((e >> 2) & 3) << 2);
      *(float4*)&L[row * 516 + kk] = *(const float4*)(base + idx);
    }
    __syncthreads();
    if (wave == 0)
      acc = wmma_rowsum16_acc(L, 0, 516, 512, 1.0f / 262144.0f, acc);
  }
  if (wave == 0) {
    const int lane = t & 31;
    if (lane == 0 || lane == 16) {
      const int rsel = (lane == 0) ? 0 : 8;
      float* dst = P3 + ((((size_t)b << 10) | (unsigned)j) << 4);
#pragma unroll
      for (int r = 0; r < 8; ++r) dst[rsel + r] = acc[r];
    }
  }
}

// Level 3 reduce stage 2: sum 256 (c1,b1,a2,a3) chunk partials -> D3
__global__ __launch_bounds__(128) void k_l3a2(const float* __restrict__ P3,
                                              float* __restrict__ D3) {
  int tt = threadIdx.x;                     // (b,yp,zp,a1)
  int b = tt >> 6;
  int yp = (tt >> 4) & 3, zp = (tt >> 2) & 3, a1 = tt & 3;
  int row = (yp << 2) | zp;
  float s = 0.f;
  for (int u = 0; u < 256; ++u) {           // (c1,b1,a2,a3)
    int chunk = ((u >> 4) << 6) | (a1 << 4) | (u & 15);
    s += P3[((((size_t)b << 10) | (unsigned)chunk) << 4) | (unsigned)row];
  }
  D3[tt] = s;
}

// Level 3 broadcast: out3[X,Y,Z] = D3[Y4][Z4][X4]
__global__ __launch_bounds__(256) void k_l3b(const float* __restrict__ D3,
                                             float* __restrict__ out) {
  __shared__ float L[64];
  const int t = threadIdx.x;
  const int tile = blockIdx.x & 4095;
  const int b = blockIdx.x >> 12;
  if (t < 64) L[t] = D3[(b << 6) | t];
  __syncthreads();
  float* ot = out + (size_t)b * 67108864u + 50331648u + ((size_t)tile << 12);
#pragma unroll
  for (int v = 0; v < 4; ++v) {
    int u = v * 256 + t;
    int Y4 = u & 3, X4 = (u >> 2) & 3;
    float4 o;
    o.x = L[(Y4 << 4) | (0 << 2) | X4];
    o.y = L[(Y4 << 4) | (1 << 2) | X4];
    o.z = L[(Y4 << 4) | (2 << 2) | X4];
    o.w = L[(Y4 << 4) | (3 << 2) | X4];
    *(float4*)(ot + (u << 2)) = o;
  }
}

extern "C" void kernel_launch(void* const* d_in, const int* in_sizes, int n_in,
                              void* d_out, int out_size, void* d_ws,
                              size_t ws_size, hipStream_t stream) {
  (void)in_sizes; (void)n_in; (void)out_size; (void)ws_size;
  const float* x = (const float*)d_in[0];
  float* out = (float*)d_out;
  float* ws = (float*)d_ws;
  float* D1 = ws;              // 524288 floats
  float* P2 = ws + 524288;     // 131072
  float* D2 = ws + 655360;     // 8192
  float* P3 = ws + 663552;     // 32768
  float* D3 = ws + 696320;     // 128   (total ~2.8 MB)

  k_l0  <<<8192, 256, 0, stream>>>(x, out);
  k_l1a <<<2048, 256, 0, stream>>>(x, D1);
  k_l1b <<<8192, 256, 0, stream>>>(D1, out);
  k_l2a <<<2048, 256, 0, stream>>>(x, P2);
  k_l2a2<<<32,   256, 0, stream>>>(P2, D2);
  k_l2b <<<8192, 256, 0, stream>>>(D2, out);
  k_l3a <<<2048, 256, 0, stream>>>(x, P3);
  k_l3a2<<<1,    128, 0, stream>>>(P3, D3);
  k_l3b <<<8192, 256, 0, stream>>>(D3, out);
}